// ChamferLoss_42021960024229
// MI455X (gfx1250) — compile-verified
//
#include <hip/hip_runtime.h>
#include <hip/hip_bf16.h>
#include <math.h>

typedef __attribute__((ext_vector_type(2))) float v2f;
typedef __attribute__((ext_vector_type(8))) float v8f;

#define BATCH 8
#define NPTS  2048   // points per set (N == M)
#define TILE  16

// d_ws layout (each BATCH*NPTS elements):
//   [0]       n1   : ||set1[b][n]||^2                    (float)
//   [1*16384] n2   : ||set2[b][m]||^2                    (float)
//   [2*16384] rmin : per-row min squared distance        (uint bits, nonneg)
//   [3*16384] cmin : per-col min squared distance        (uint bits, nonneg)

__global__ void chamfer_norms_init(const float* __restrict__ set1,
                                   const float* __restrict__ set2,
                                   float* __restrict__ n1,
                                   float* __restrict__ n2,
                                   unsigned int* __restrict__ rmin,
                                   unsigned int* __restrict__ cmin) {
    int idx = blockIdx.x * blockDim.x + threadIdx.x;   // 0 .. BATCH*NPTS-1
    if (idx >= BATCH * NPTS) return;
    const float* p1 = set1 + (size_t)idx * 3;
    const float* p2 = set2 + (size_t)idx * 3;
    float a0 = p1[0], a1 = p1[1], a2 = p1[2];
    float b0 = p2[0], b1 = p2[1], b2 = p2[2];
    n1[idx] = a0 * a0 + a1 * a1 + a2 * a2;
    n2[idx] = b0 * b0 + b1 * b1 + b2 * b2;
    rmin[idx] = 0x7F800000u;   // +inf
    cmin[idx] = 0x7F800000u;
}

// One block = (batch, 16-row strip). 8 waves; each wave covers 256 columns
// (16 column tiles), one v_wmma_f32_16x16x4_f32 per 16x16 tile.
//
// Math per tile: D = A x B + C with
//   A[m][0..2] = x_m,          A[m][3] = 1
//   B[0..2][n] = -2 * y_n,     B[3][n] = ||y_n||^2
//   C[m][n]    = ||x_m||^2
// => D[m][n] = ||x_m||^2 - 2 x_m.y_n + ||y_n||^2  (full squared distance
//    straight out of the matrix unit).
//
// All min reductions run in the raw float domain (tiny negatives from FP
// cancellation are fine for fminf); we clamp to >= 0 only at the points
// where values enter the uint-bit-ordering domain (the atomicMin's).
__global__ void __launch_bounds__(256)
chamfer_tiles(const float* __restrict__ set1,
              const float* __restrict__ set2,
              const float* __restrict__ n1,
              const float* __restrict__ n2,
              unsigned int* __restrict__ rmin,
              unsigned int* __restrict__ cmin) {
    const int b       = blockIdx.y;
    const int rowBase = blockIdx.x * TILE;
    const int wave    = threadIdx.x >> 5;   // 0..7
    const int lane    = threadIdx.x & 31;
    const int half    = lane >> 4;          // 0: holds K=0,1 ; 1: holds K=2,3
    const int l       = lane & 15;

    // ---- A fragment (branchless): both halves load the full point ----
    // ISA 16x4 A layout: lanes 0-15 -> {K=0, K=1} of row M=lane,
    //                    lanes 16-31 -> {K=2, K=3} of row M=lane-16.
    const float* xr = set1 + ((size_t)b * NPTS + rowBase + l) * 3;
    const float x0 = xr[0], x1 = xr[1], x2 = xr[2];
    v2f afrag;
    afrag.x = half ? x2   : x0;
    afrag.y = half ? 1.0f : x1;     // K=3 pad = 1 -> picks up ||y||^2 from B

    // ---- C fragment: row norms, slot v holds row M = v + 8*half ----
    const float* n1p = n1 + (size_t)b * NPTS + rowBase + 8 * half;
    v8f cfrag;
#pragma unroll
    for (int v = 0; v < 8; ++v) cfrag[v] = n1p[v];

    const float INF = __builtin_inff();
    v8f rminAcc;
#pragma unroll
    for (int v = 0; v < 8; ++v) rminAcc[v] = INF;

    const float* s2b = set2 + (size_t)b * NPTS * 3;
    const float* n2b = n2 + (size_t)b * NPTS;

    const int colStart = wave * 256;        // this wave's 256-column slice
#pragma unroll 4
    for (int t = 0; t < 16; ++t) {
        const int n = colStart + t * TILE + l;   // same column for both halves

        // ---- B fragment (branchless) ----
        const float* yc = s2b + (size_t)n * 3;
        const float y0 = yc[0], y1 = yc[1], y2 = yc[2];
        const float cn = n2b[n];
        v2f bfrag;
        bfrag.x = half ? (-2.0f * y2) : (-2.0f * y0);
        bfrag.y = half ? cn           : (-2.0f * y1);

        // D[m][n] = squared distance (raw; may be tiny-negative)
        v8f d = __builtin_amdgcn_wmma_f32_16x16x4_f32(
            /*neg_a=*/false, afrag, /*neg_b=*/false, bfrag,
            /*c_mod=*/(short)0, cfrag, /*reuse_a=*/false, /*reuse_b=*/false);

        // running row mins (raw domain)
#pragma unroll
        for (int v = 0; v < 8; ++v) rminAcc[v] = fminf(rminAcc[v], d[v]);

        // per-tile column min over this half's 8 rows (pairwise tree)
        float m01 = fminf(d[0], d[1]);
        float m23 = fminf(d[2], d[3]);
        float m45 = fminf(d[4], d[5]);
        float m67 = fminf(d[6], d[7]);
        float colAcc = fminf(fminf(m01, m23), fminf(m45, m67));

        // combine M=v (half 0) with M=v+8 (half 1): same column n
        colAcc = fminf(colAcc, __shfl_xor(colAcc, 16, 32));
        colAcc = fmaxf(colAcc, 0.0f);       // clamp once, entering uint domain
        if (half == 0)
            atomicMin(&cmin[(size_t)b * NPTS + n], __float_as_uint(colAcc));
    }

    // ---- row-min reduction across the 16 lanes of each half ----
#pragma unroll
    for (int v = 0; v < 8; ++v) {
        float r = rminAcc[v];
        r = fminf(r, __shfl_xor(r, 1, 32));
        r = fminf(r, __shfl_xor(r, 2, 32));
        r = fminf(r, __shfl_xor(r, 4, 32));
        r = fminf(r, __shfl_xor(r, 8, 32));
        rminAcc[v] = fmaxf(r, 0.0f);        // clamp once, entering uint domain
    }
    if (l == 0) {
        unsigned int* rp = rmin + (size_t)b * NPTS + rowBase + 8 * half;
#pragma unroll
        for (int v = 0; v < 8; ++v)
            atomicMin(&rp[v], __float_as_uint(rminAcc[v]));
    }
}

// One block per batch: loss[b] = mean(sqrt(rmin)) + mean(sqrt(cmin))
__global__ void __launch_bounds__(256)
chamfer_finalize(const unsigned int* __restrict__ rmin,
                 const unsigned int* __restrict__ cmin,
                 float* __restrict__ out) {
    const int b   = blockIdx.x;
    const int tid = threadIdx.x;
    __shared__ float red[256];

    float s1 = 0.0f, s2 = 0.0f;
    for (int i = tid; i < NPTS; i += 256) {
        s1 += sqrtf(__uint_as_float(rmin[(size_t)b * NPTS + i]));
        s2 += sqrtf(__uint_as_float(cmin[(size_t)b * NPTS + i]));
    }
    red[tid] = s1 + s2;   // same 1/NPTS factor for both means
    __syncthreads();
    for (int off = 128; off > 0; off >>= 1) {
        if (tid < off) red[tid] += red[tid + off];
        __syncthreads();
    }
    if (tid == 0) out[b] = red[0] * (1.0f / (float)NPTS);
}

extern "C" void kernel_launch(void* const* d_in, const int* in_sizes, int n_in,
                              void* d_out, int out_size, void* d_ws, size_t ws_size,
                              hipStream_t stream) {
    const float* set1 = (const float*)d_in[0];   // [8, 2048, 3] fp32
    const float* set2 = (const float*)d_in[1];   // [8, 2048, 3] fp32
    float* out = (float*)d_out;                  // [8] fp32

    const size_t E = (size_t)BATCH * NPTS;       // 16384 elements per array
    float*        n1   = (float*)d_ws;
    float*        n2   = n1 + E;
    unsigned int* rmin = (unsigned int*)(n2 + E);
    unsigned int* cmin = rmin + E;

    chamfer_norms_init<<<(int)((E + 255) / 256), 256, 0, stream>>>(
        set1, set2, n1, n2, rmin, cmin);

    dim3 grid(NPTS / TILE, BATCH);               // 128 row strips x 8 batches
    chamfer_tiles<<<grid, 256, 0, stream>>>(set1, set2, n1, n2, rmin, cmin);

    chamfer_finalize<<<BATCH, 256, 0, stream>>>(rmin, cmin, out);
}